// SidedDistance_14482629722267
// MI455X (gfx1250) — compile-verified
//
#include <hip/hip_runtime.h>

// Nearest-neighbor argmin via V_WMMA_F32_16X16X4_F32 on gfx1250 (wave32).
//
//   d2'[n][m] = -2 * S1[n] . S2[m] + ||S2[m]||^2      (argmin-equivalent to d2)
// computed as a 16x16x4 fp32 WMMA with A-row = (x,y,z,1), B-col = (-2x,-2y,-2z,|p|^2).

typedef float v2f __attribute__((ext_vector_type(2)));
typedef float v8f __attribute__((ext_vector_type(8)));

#define BLOCK_THREADS 256   // 8 waves of 32
#define ROWS_PER_WAVE 16    // one 16-row WMMA strip of S1 per wave
#define ROWS_PER_BLOCK 128  // 8 waves * 16 rows
#define TILE_M 256          // S2 points staged in LDS per outer iteration (4 KB)

__global__ __launch_bounds__(BLOCK_THREADS)
void nn_argmin_wmma_f32(const float* __restrict__ S1,
                        const float* __restrict__ S2,
                        long long* __restrict__ out,
                        int N, int M)
{
    __shared__ float4 sq[TILE_M];   // (-2x, -2y, -2z, ||p||^2) per staged S2 point

    const int tid  = threadIdx.x;
    const int lane = tid & 31;
    const int wave = tid >> 5;
    const int li   = lane & 15;     // lane-in-half  -> column / A-row selector
    const int half = lane >> 4;     // 0: K={0,1} / rows 0-7 ; 1: K={2,3} / rows 8-15

    const int blocksPerBatch = N / ROWS_PER_BLOCK;
    const int b         = blockIdx.x / blocksPerBatch;
    const int rowBase   = (blockIdx.x % blocksPerBatch) * ROWS_PER_BLOCK;
    const int stripBase = rowBase + wave * ROWS_PER_WAVE;   // first S1 row of this wave

    // ---- A fragment: 16x4 fp32 (ISA 7.12.2). Lane half 0 holds K=0,1; half 1 K=2,3.
    const float* p1 = S1 + ((size_t)b * N + stripBase + li) * 3;
    const float ax0 = p1[0], ay0 = p1[1], az0 = p1[2];
    v2f afrag;
    afrag.x = half ? az0 : ax0;     // K = 2*half
    afrag.y = half ? 1.0f : ay0;    // K = 2*half + 1   (w-pad = 1 carries ||b||^2 term)

    float minv[8];
    int   mini[8];
#pragma unroll
    for (int v = 0; v < 8; ++v) { minv[v] = 3.4e38f; mini[v] = 0; }

    for (int mt = 0; mt < M; mt += TILE_M) {
        __syncthreads();    // protect LDS tile before overwrite
        {
            // Cooperative stage: thread t preprocesses S2 point (mt + t).
            const float* p2 = S2 + ((size_t)b * M + mt + tid) * 3;
            const float x = p2[0], y = p2[1], z = p2[2];
            sq[tid] = make_float4(-2.0f * x, -2.0f * y, -2.0f * z,
                                  x * x + y * y + z * z);
            // Speculative prefetch of the next tile (global_prefetch_b8).
            const int mn = (mt + TILE_M < M) ? (mt + TILE_M) : 0;
            __builtin_prefetch(S2 + ((size_t)b * M + mn + tid) * 3, 0, 0);
        }
        __syncthreads();

#pragma unroll 4
        for (int t = 0; t < TILE_M; t += 16) {
            // ---- B fragment: 4x16 fp32, row-striped across lanes; half selects K pair.
            const float4 q = sq[t + li];
            v2f bfrag;
            bfrag.x = half ? q.z : q.x;   // K = 2*half      (-2z | -2x)
            bfrag.y = half ? q.w : q.y;   // K = 2*half + 1  (|p|^2 | -2y)

            v8f c = {};
            // D = A(16x4) * B(4x16) + 0 ; full fp32, one 16x16 distance tile.
            v8f d = __builtin_amdgcn_wmma_f32_16x16x4_f32(
                /*neg_a=*/false, afrag, /*neg_b=*/false, bfrag,
                /*c_mod=*/(short)0, c, /*reuse_a=*/false, /*reuse_b=*/false);

            const int col = mt + t + li;  // column index is per-lane, fixed over v
#pragma unroll
            for (int v = 0; v < 8; ++v) {
                if (d[v] < minv[v]) { minv[v] = d[v]; mini[v] = col; }
            }
        }
    }

    // ---- Argmin reduction across the 16 lanes of each half (wave32 butterfly).
    // VGPR v / lane-half h holds D-row M = v + 8*h (ISA 7.12.2 C/D layout).
#pragma unroll
    for (int v = 0; v < 8; ++v) {
        float val = minv[v];
        int   idx = mini[v];
#pragma unroll
        for (int o = 8; o >= 1; o >>= 1) {   // xor masks 8,4,2,1 stay inside the half
            const float ov = __shfl_xor(val, o, 32);
            const int   oi = __shfl_xor(idx, o, 32);
            if (ov < val || (ov == val && oi < idx)) { val = ov; idx = oi; }
        }
        if (li == 0) {
            out[(size_t)b * N + stripBase + v + 8 * half] = (long long)idx;
        }
    }
}

extern "C" void kernel_launch(void* const* d_in, const int* in_sizes, int n_in,
                              void* d_out, int out_size, void* d_ws, size_t ws_size,
                              hipStream_t stream) {
    (void)in_sizes; (void)n_in; (void)d_ws; (void)ws_size; (void)out_size;
    const float* S1 = (const float*)d_in[0];   // (4, 8192, 3) f32
    const float* S2 = (const float*)d_in[1];   // (4, 8192, 3) f32
    long long*  out = (long long*)d_out;       // (4, 8192) int64

    const int B = 4, N = 8192, M = 8192;
    dim3 grid(B * (N / ROWS_PER_BLOCK));       // 256 blocks
    dim3 block(BLOCK_THREADS);                 // 8 waves
    nn_argmin_wmma_f32<<<grid, block, 0, stream>>>(S1, S2, out, N, M);
}